// DA_conv_6038724018987
// MI455X (gfx1250) — compile-verified
//
#include <hip/hip_runtime.h>

typedef float v2f __attribute__((ext_vector_type(2)));
typedef float v8f __attribute__((ext_vector_type(8)));

#define Bn 16
#define Cn 64
#define Hn 128
#define Wn 128
#define Rn 4
#define TW 32
#define GWS 578   // guide-weight row stride (even -> b64 A loads; %64==2 -> no bank conflicts)

__device__ __forceinline__ float leakyf(float x) { return x > 0.f ? x : 0.1f * x; }

// ws layout (floats): [0,256) h1 | [256,512) a1 | [512,512+36864) kern(B,R,C,9) | then att(B,R,C) 4096

__global__ void hyper_small(const float* __restrict__ x_deg,
                            const float* __restrict__ kw1,
                            const float* __restrict__ ca_w1,
                            float* __restrict__ ws)
{
    int t = threadIdx.x;            // 0..511
    int sel = t >> 8;               // 0: h1, 1: a1
    int j = t & 255;
    int b = j >> 4, k = j & 15;
    const float* w = (sel ? ca_w1 : kw1) + k * Cn;
    const float* x = x_deg + b * Cn;
    float s = 0.f;
    #pragma unroll 8
    for (int c = 0; c < Cn; ++c) s += x[c] * w[c];
    ws[sel * 256 + j] = leakyf(s);
}

__global__ void hyper_expand(const float* __restrict__ kw2,
                             const float* __restrict__ ca_w2,
                             float* __restrict__ ws)
{
    const float* h1 = ws;
    const float* a1 = ws + 256;
    float* kern = ws + 512;
    float* att  = ws + 512 + Bn * Rn * Cn * 9;
    const int NK = Bn * Rn * Cn * 9;          // 36864
    int i = blockIdx.x * blockDim.x + threadIdx.x;
    if (i < NK) {
        int b = i / (Rn * Cn * 9);
        int g = (i / (Cn * 9)) % Rn;
        int o = i % (Cn * 9);
        const float* hv = h1 + b * 16 + g * 4;
        const float* wv = kw2 + (g * (Cn * 9) + o) * 4;
        kern[i] = hv[0]*wv[0] + hv[1]*wv[1] + hv[2]*wv[2] + hv[3]*wv[3];
    } else if (i < NK + Bn * Rn * Cn) {
        int j = i - NK;
        int b = j / (Rn * Cn);
        int g = (j / Cn) % Rn;
        int o = j % Cn;
        const float* av = a1 + b * 16 + g * 4;
        const float* wv = ca_w2 + (g * Cn + o) * 4;
        float s = av[0]*wv[0] + av[1]*wv[1] + av[2]*wv[2] + av[3]*wv[3];
        att[j] = 1.f / (1.f + __expf(-s));
    }
}

__global__ __launch_bounds__(256)
void fused_main(const float* __restrict__ x0,
                const float* __restrict__ conv_w,
                const float* __restrict__ conv_b,
                const float* __restrict__ g1_w,
                const float* __restrict__ g1_b,
                const float* __restrict__ g2_w,
                const float* __restrict__ g2_b,
                const float* __restrict__ kern_all,
                const float* __restrict__ att_all,
                float* __restrict__ out)
{
    __shared__ float sx[Cn * 3 * 34];          // x0 halo tile                26112 B
    __shared__ float sgw[9 * GWS];             // guide W (8 rows + zero row) 20808 B (skern overlay)
    __shared__ float satt[Rn * Cn];            //                              1024 B
    __shared__ float scr[8 * 16 * 17];         // guide partials / st tile     8704 B (overlay)
    __shared__ float sguide[8 * 32];           //                              1024 B
    __shared__ float sconvb[Cn];               //                               256 B
    __shared__ int sr1[TW], sr2[TW];           //                               256 B
    __shared__ unsigned int koffb[576];        // im2col BYTE-offset LUT       2304 B

    const int tid = threadIdx.x;
    const int w0 = blockIdx.x * TW;
    const int h  = blockIdx.y;
    const int b  = blockIdx.z;

    // ---- phase 1: stage x0 halo tile, guide weights (+zero row), att, bias, im2col LUT
    for (int i = tid; i < Cn * 3 * 34; i += 256) {
        int c = i / 102;
        int rem = i - c * 102;
        int row = rem / 34;
        int col = rem - row * 34;
        int gh = h - 1 + row;
        int gw = w0 - 1 + col;
        float v = 0.f;
        if ((unsigned)gh < (unsigned)Hn && (unsigned)gw < (unsigned)Wn)
            v = x0[((b * Cn + c) * Hn + gh) * Wn + gw];
        sx[i] = v;
    }
    for (int i = tid; i < 8 * 576; i += 256) {
        int row = i / 576;
        int col = i - row * 576;
        float v = (row < 4) ? g1_w[row * 576 + col] : g2_w[(row - 4) * 576 + col];
        sgw[row * GWS + col] = v;
    }
    for (int i = tid; i < GWS; i += 256) sgw[8 * GWS + i] = 0.f;   // zero row for A rows 8..15
    for (int i = tid; i < Rn * Cn; i += 256) satt[i] = att_all[b * Rn * Cn + i];
    if (tid < Cn) sconvb[tid] = conv_b[tid];
    for (int k = tid; k < 576; k += 256) {
        int c = k / 9;
        int t = k - c * 9;
        int kh = t / 3;
        int kw = t - kh * 3;
        koffb[k] = (unsigned int)((c * 102 + kh * 34 + kw) * 4);   // byte offsets
    }
    __syncthreads();

    // ---- phase 2: guide convs as WMMA GEMM  G[16x32] = W[16x576] * im2col[576x32]
    // rows 0..3 = guide1 r, rows 4..7 = guide2 r, rows 8..15 zero (via zero row).
    // 8 waves = 2 N-tiles x 4 K-chunks (144 each); partials -> scr (stride-17 tiles)
    {
        int wave  = tid >> 5;
        int lane  = tid & 31;
        int ntile = wave >> 2;       // 0..1
        int chunk = wave & 3;        // 0..3
        int l15 = lane & 15;
        int hi  = lane >> 4;
        const char* sxb = (const char*)sx + (ntile * 16 + l15) * 4;  // per-lane pixel base
        const float* arow = sgw + ((l15 < 8) ? l15 : 8) * GWS;       // lanes 8..15 -> zero row
        v8f acc = {};
        int kb = chunk * 144 + 2 * hi;   // A VGPR0 holds K0/K2, VGPR1 K1/K3; kb even
        #pragma unroll 4
        for (int s = 0; s < 36; ++s, kb += 4) {
            v2f A, Bv;
            A.x  = arow[kb];             // contiguous, 8B aligned -> b64 A loads
            A.y  = arow[kb + 1];
            Bv.x = *(const float*)(sxb + koffb[kb]);      // single v_add per element
            Bv.y = *(const float*)(sxb + koffb[kb + 1]);
            acc = __builtin_amdgcn_wmma_f32_16x16x4_f32(
                false, A, false, Bv, (short)0, acc, false, false);
        }
        float* pw = scr + wave * 272;    // 16x17 padded tile
        #pragma unroll
        for (int r = 0; r < 8; ++r)
            pw[(r + 8 * hi) * 17 + l15] = acc[r];
    }
    __syncthreads();

    // ---- phase 3: reduce 4 K-chunk partials (+bias) -> sguide ; stage kern[b] into sgw overlay
    {
        int nt = tid >> 7;           // N-tile 0..1
        int m  = (tid >> 4) & 7;     // guide row 0..7
        int nn = tid & 15;
        float s = 0.f;
        #pragma unroll
        for (int ch = 0; ch < 4; ++ch)
            s += scr[(nt * 4 + ch) * 272 + m * 17 + nn];
        s += (m < 4) ? g1_b[m] : g2_b[m - 4];
        sguide[m * 32 + nt * 16 + nn] = s;
    }
    {
        float* skern = sgw;          // guide weights dead; reuse region
        for (int i = tid; i < Rn * Cn * 9; i += 256)
            skern[i] = kern_all[b * Rn * Cn * 9 + i];
    }
    __syncthreads();

    // ---- phase 4: per-pixel argmax over R (first-max wins, matching jnp.argmax)
    if (tid < 64) {
        int p = tid & 31;
        int g = tid >> 5;
        const float* gv = sguide + g * 128;
        float best = gv[p];
        int bi = 0;
        #pragma unroll
        for (int r = 1; r < 4; ++r) {
            float v = gv[r * 32 + p];
            if (v > best) { best = v; bi = r; }
        }
        if (g == 0) sr1[p] = bi; else sr2[p] = bi;
    }
    __syncthreads();

    // ---- phase 5: depthwise 3x3 with routed region kernel + leaky -> scr[c*33+p]
    {
        const float* skern = sgw;
        for (int i = tid; i < Cn * TW; i += 256) {
            int c = i >> 5;
            int p = i & 31;
            const float* kr = skern + (sr1[p] * Cn + c) * 9;
            const float* xc = sx + c * 102;
            float s = 0.f;
            #pragma unroll
            for (int kh = 0; kh < 3; ++kh)
                s += kr[kh*3+0] * xc[kh*34 + p]
                   + kr[kh*3+1] * xc[kh*34 + p + 1]
                   + kr[kh*3+2] * xc[kh*34 + p + 2];
            scr[c * 33 + p] = leakyf(s);
        }
    }
    __syncthreads();

    // ---- phase 6: 1x1 conv as f32 WMMA GEMM  D[o,p] = sum_c conv_w[o,c] * t[c,p]
    {
        int wave  = tid >> 5;
        int lane  = tid & 31;
        int mbase = (wave & 3) * 16;
        int nbase = (wave >> 2) * 16;
        int l15 = lane & 15;
        int hi  = lane >> 4;
        v8f acc = {};
        #pragma unroll
        for (int kb = 0; kb < Cn; kb += 4) {
            int k0 = kb + 2 * hi;
            v2f A, Bv;
            A.x  = conv_w[(mbase + l15) * Cn + k0];
            A.y  = conv_w[(mbase + l15) * Cn + k0 + 1];
            Bv.x = scr[k0 * 33 + nbase + l15];
            Bv.y = scr[(k0 + 1) * 33 + nbase + l15];
            acc = __builtin_amdgcn_wmma_f32_16x16x4_f32(
                false, A, false, Bv, (short)0, acc, false, false);
        }
        // epilogue: + conv_b + CA branch (x0 * att[r2]); single fused output write
        #pragma unroll
        for (int r = 0; r < 8; ++r) {
            int o = mbase + r + 8 * hi;       // D: VGPR r -> M=r (lanes<16) / r+8
            int n = nbase + l15;
            float xc  = sx[o * 102 + 34 + n + 1];    // x0[b,o,h,w0+n] center row
            float val = acc[r] + sconvb[o] + xc * satt[sr2[n] * Cn + o];
            out[((b * Cn + o) * Hn + h) * Wn + w0 + n] = val;
        }
    }
}

extern "C" void kernel_launch(void* const* d_in, const int* in_sizes, int n_in,
                              void* d_out, int out_size, void* d_ws, size_t ws_size,
                              hipStream_t stream) {
    const float* x0     = (const float*)d_in[0];
    const float* x_deg  = (const float*)d_in[1];
    const float* kw1    = (const float*)d_in[2];
    const float* kw2    = (const float*)d_in[3];
    const float* conv_w = (const float*)d_in[4];
    const float* conv_b = (const float*)d_in[5];
    const float* ca_w1  = (const float*)d_in[6];
    const float* ca_w2  = (const float*)d_in[7];
    const float* g1_w   = (const float*)d_in[8];
    const float* g1_b   = (const float*)d_in[9];
    const float* g2_w   = (const float*)d_in[10];
    const float* g2_b   = (const float*)d_in[11];
    float* ws  = (float*)d_ws;
    float* out = (float*)d_out;

    hipLaunchKernelGGL(hyper_small, dim3(1), dim3(512), 0, stream, x_deg, kw1, ca_w1, ws);
    hipLaunchKernelGGL(hyper_expand, dim3(160), dim3(256), 0, stream, kw2, ca_w2, ws);

    const float* kern_all = ws + 512;
    const float* att_all  = ws + 512 + Bn * Rn * Cn * 9;
    hipLaunchKernelGGL(fused_main, dim3(Wn / TW, Hn, Bn), dim3(256), 0, stream,
                       x0, conv_w, conv_b, g1_w, g1_b, g2_w, g2_b,
                       kern_all, att_all, out);
}